// Apply2DTform_66838281060466
// MI455X (gfx1250) — compile-verified
//
#include <hip/hip_runtime.h>

// Apply2DTform: out[b,i,j] = 4-tap round-based interpolation of Img[b] at the
// affine-mapped coordinate of output pixel (i,j).
//   Img:   [32, 512, 512, 1] f32
//   Tform: [32, 6]            f32  (m00 m01 m10 m11 v0 v1)
//   out:   [32, 512, 512, 1] f32
//
// Memory-bound gather kernel. Each wave32 thread produces 4 contiguous output
// pixels (one global_store_b128, non-temporal). Batch index is blockIdx.y so
// the transform coefficients are scalar (SGPR) loads.

#define IMG_H 512
#define IMG_W 512
#define OUT_H 512
#define OUT_W 512
#define PIX_PER_THREAD 4

typedef float v4f __attribute__((ext_vector_type(4)));

__global__ __launch_bounds__(256, 8)
void apply2dtform_kernel(const float* __restrict__ img,    // [B,512,512]
                         const float* __restrict__ tform,  // [B,6]
                         float* __restrict__ out)          // [B,512,512]
{
    const int b    = blockIdx.y;                                   // uniform -> SGPR
    const int pix0 = (blockIdx.x * blockDim.x + threadIdx.x) * PIX_PER_THREAD;
    const int i    = pix0 >> 9;           // row   (512 cols per row)
    const int j0   = pix0 & (OUT_W - 1);  // first of 4 consecutive cols

    // Transform coefficients: uniform across the block -> s_load_b*.
    const float* __restrict__ tf = tform + b * 6;
    const float m00 = tf[0], m01 = tf[1], m10 = tf[2], m11 = tf[3];
    const float v0  = tf[4], v1  = tf[5];

    // ax[i] = linspace(-1,1,512)[i] = (2*i - 511)/511  (single rounding, exact
    // endpoints; matters because round-based weights are discontinuous at .5).
    const float axi = (float)(2 * i - 511) * (1.0f / 511.0f);

    const float* __restrict__ imgb = img + ((size_t)b << 18);      // b*512*512

    v4f res;

#pragma unroll
    for (int k = 0; k < PIX_PER_THREAD; ++k) {
        const int   j   = j0 + k;
        const float ayj = (float)(2 * j - 511) * (1.0f / 511.0f);

        // Source coordinates: bg = M*[ax;ay] + V, then rescale to pixel space
        // x = 0.5*(xs+1)*(513-1-1) = 0.5*(xs+1)*511
        const float xs = fmaf(m00, axi, fmaf(m01, ayj, v0));
        const float ys = fmaf(m10, axi, fmaf(m11, ayj, v1));
        const float x  = 0.5f * (xs + 1.0f) * 511.0f;
        const float y  = 0.5f * (ys + 1.0f) * 511.0f;

        // jnp.round == round-half-to-even == rintf under default RNE mode.
        const int xr = (int)rintf(x);
        const int yr = (int)rintf(y);

        // Clip to padded-image range [0, 512].
        const int x0 = min(max(xr,     0), IMG_H);
        const int x1 = min(max(xr + 1, 0), IMG_H);
        const int y0 = min(max(yr,     0), IMG_W);
        const int y1 = min(max(yr + 1, 0), IMG_W);

        // Zero-padded gathers: index 512 in either dim lands in the pad row/col.
        const float I00 = (x0 < IMG_H && y0 < IMG_W) ? imgb[x0 * IMG_W + y0] : 0.0f;
        const float I01 = (x0 < IMG_H && y1 < IMG_W) ? imgb[x0 * IMG_W + y1] : 0.0f;
        const float I10 = (x1 < IMG_H && y0 < IMG_W) ? imgb[x1 * IMG_W + y0] : 0.0f;
        const float I11 = (x1 < IMG_H && y1 < IMG_W) ? imgb[x1 * IMG_W + y1] : 0.0f;

        // Weights from the *clipped* integer coords (matches reference; can be
        // negative / >1 because of round + clipping).
        const float x0f = (float)x0, x1f = (float)x1;
        const float y0f = (float)y0, y1f = (float)y1;
        const float wx1 = x1f - x, wx0 = x - x0f;
        const float wy1 = y1f - y, wy0 = y - y0f;

        float acc = (wx1 * wy1) * I00;
        acc = fmaf(wx1 * wy0, I01, acc);
        acc = fmaf(wx0 * wy1, I10, acc);
        acc = fmaf(wx0 * wy0, I11, acc);
        res[k] = acc;
    }

    // One-shot streaming output: non-temporal b128 store keeps L2 capacity
    // for the gathered image data (all 32 images fit in the 192MB L2).
    __builtin_nontemporal_store(res, reinterpret_cast<v4f*>(
        out + ((size_t)b << 18) + pix0));
}

extern "C" void kernel_launch(void* const* d_in, const int* in_sizes, int n_in,
                              void* d_out, int out_size, void* d_ws, size_t ws_size,
                              hipStream_t stream) {
    const float* img   = (const float*)d_in[0];   // [32,512,512,1] f32
    const float* tform = (const float*)d_in[1];   // [32,6]         f32
    float*       outp  = (float*)d_out;           // [32,512,512,1] f32

    const int B = in_sizes[1] / 6;                // 32
    // Each block: 256 threads x 4 pixels = 1024 contiguous pixels (2 rows).
    dim3 block(256, 1, 1);
    dim3 grid((OUT_H * OUT_W) / (256 * PIX_PER_THREAD), B, 1);
    apply2dtform_kernel<<<grid, block, 0, stream>>>(img, tform, outp);
}